// AE_63548336111686
// MI455X (gfx1250) — compile-verified
//
#include <hip/hip_runtime.h>
#include <math.h>

// ---------------------------------------------------------------------------
// LISTA unrolled encoder on MI455X (gfx1250), fp32 via V_WMMA_F32_16X16X4_F32.
// Problem: M=512, P=2048, B=4096, 30 layers.
//   S   = I - 0.1 * W^T W            [P,P]  (symmetric)
//   WTx = 0.1 * X W                  [B,P]
//   Z   <- soft(Z S + WTx)  x30      [B,P]
//   xhat = Z W^T                     [B,M]
// Working set (~112MB) fits in the 192MB L2 -> compute bound on the f32
// matrix pipe. One tiled WMMA GEMM template; double-buffered LDS staging with
// register prefetch so global-load latency hides behind the 64 WMMAs/k-block.
// ---------------------------------------------------------------------------

typedef float v2f __attribute__((ext_vector_type(2)));
typedef float v8f __attribute__((ext_vector_type(8)));

static constexpr int Mdim = 512;
static constexpr int Pdim = 2048;
static constexpr int Bdim = 4096;
static constexpr int NUM_LAYERS = 30;
static constexpr float STEP = 0.1f;
static constexpr float THRESH = 0.01f;   // LAM * STEP

#define BM 128
#define BN 128
#define BK 32
#define AS 36   // A LDS row stride (floats): 16B-aligned rows, conflict-free frags
#define BS 34   // B LDS row stride (floats): 8B-aligned rows, conflict-free frags

enum { EPI_NONE = 0, EPI_SCALE = 1, EPI_SOFT = 2, EPI_S = 3 };

__device__ __forceinline__ float soft_thresh(float pre) {
    float a = fabsf(pre) - THRESH;
    return a > 0.0f ? copysignf(a, pre) : 0.0f;
}

// ---- staging helpers: global -> registers -> LDS ---------------------------
template <int TA>
__device__ __forceinline__ void ld_a(const float* __restrict__ A, int lda,
                                     int m0, int k0, int tid, float4 (&r)[4]) {
    if (TA == 0) {               // Atile[i][k] = A[(m0+i)*lda + k0+k]
        const int kq = (tid & 7) * 4, ib = tid >> 3;
#pragma unroll
        for (int it = 0; it < 4; ++it)
            r[it] = *(const float4*)&A[(size_t)(m0 + ib + it * 32) * lda + k0 + kq];
    } else {                     // Atile[i][k] = A[(k0+k)*lda + m0+i]  (W^T)
        const int i4 = (tid & 31) * 4, kb = tid >> 5;
#pragma unroll
        for (int it = 0; it < 4; ++it)
            r[it] = *(const float4*)&A[(size_t)(k0 + kb + it * 8) * lda + m0 + i4];
    }
}

template <int TA>
__device__ __forceinline__ void st_a(float* Al, int tid, const float4 (&r)[4]) {
    if (TA == 0) {
        const int kq = (tid & 7) * 4, ib = tid >> 3;
#pragma unroll
        for (int it = 0; it < 4; ++it)
            *(float4*)&Al[(ib + it * 32) * AS + kq] = r[it];   // rows 16B aligned
    } else {
        const int i4 = (tid & 31) * 4, kb = tid >> 5;
#pragma unroll
        for (int it = 0; it < 4; ++it) {
            const int k = kb + it * 8;
            Al[(i4 + 0) * AS + k] = r[it].x;
            Al[(i4 + 1) * AS + k] = r[it].y;
            Al[(i4 + 2) * AS + k] = r[it].z;
            Al[(i4 + 3) * AS + k] = r[it].w;
        }
    }
}

template <int TB>
__device__ __forceinline__ void ld_b(const float* __restrict__ Bm, int ldb,
                                     int n0, int k0, int tid, float4 (&r)[4]) {
    if (TB == 0) {               // Btile[n][k] = B[(k0+k)*ldb + n0+n]
        const int n4 = (tid & 31) * 4, kb = tid >> 5;
#pragma unroll
        for (int it = 0; it < 4; ++it)
            r[it] = *(const float4*)&Bm[(size_t)(k0 + kb + it * 8) * ldb + n0 + n4];
    } else {                     // Btile[n][k] = B[(n0+n)*ldb + k0+k]  (W^T)
        const int kq = (tid & 7) * 4, nb = tid >> 3;
#pragma unroll
        for (int it = 0; it < 4; ++it)
            r[it] = *(const float4*)&Bm[(size_t)(n0 + nb + it * 32) * ldb + k0 + kq];
    }
}

template <int TB>
__device__ __forceinline__ void st_b(float* Bl, int tid, const float4 (&r)[4]) {
    if (TB == 0) {
        const int n4 = (tid & 31) * 4, kb = tid >> 5;
#pragma unroll
        for (int it = 0; it < 4; ++it) {
            const int k = kb + it * 8;
            Bl[(n4 + 0) * BS + k] = r[it].x;
            Bl[(n4 + 1) * BS + k] = r[it].y;
            Bl[(n4 + 2) * BS + k] = r[it].z;
            Bl[(n4 + 3) * BS + k] = r[it].w;
        }
    } else {
        const int kq = (tid & 7) * 4, nb = tid >> 3;
#pragma unroll
        for (int it = 0; it < 4; ++it) {
            const int n = nb + it * 32;
            *(float2*)&Bl[n * BS + kq]     = make_float2(r[it].x, r[it].y); // 8B ok
            *(float2*)&Bl[n * BS + kq + 2] = make_float2(r[it].z, r[it].w);
        }
    }
}

// C[MxN] = epilogue( A_op @ B_op ), K-reduction.
template <int TA, int TB, int EPI>
__global__ __launch_bounds__(256) void gemm_wmma(
    const float* __restrict__ A, int lda,
    const float* __restrict__ Bm, int ldb,
    float* __restrict__ C, int ldc,
    const float* __restrict__ extra,   // WTx for EPI_SOFT, indexed [row*ldc+col]
    int K)
{
    __shared__ __align__(16) float Al[2][BM * AS];
    __shared__ __align__(16) float Bl[2][BN * BS];

    const int tid  = threadIdx.x;
    const int lane = tid & 31;
    const int wid  = tid >> 5;
    const int wm   = (wid & 1) * 64;   // wave row offset   (2 waves in M)
    const int wn   = (wid >> 1) * 32;  // wave col offset   (4 waves in N)
    const int half = lane >> 4;        // 0: K={0,1}, 1: K={2,3} (f32 WMMA layout)
    const int l15  = lane & 15;
    const int m0   = blockIdx.y * BM;
    const int n0   = blockIdx.x * BN;

    v8f acc[4][2];
#pragma unroll
    for (int mi = 0; mi < 4; ++mi)
#pragma unroll
        for (int ni = 0; ni < 2; ++ni)
#pragma unroll
            for (int r = 0; r < 8; ++r) acc[mi][ni][r] = 0.0f;

    float4 ar[4], br[4];

    // prologue: stage k-block 0 into buffer 0
    ld_a<TA>(A, lda, m0, 0, tid, ar);
    ld_b<TB>(Bm, ldb, n0, 0, tid, br);
    st_a<TA>(Al[0], tid, ar);
    st_b<TB>(Bl[0], tid, br);
    __syncthreads();

    const int nkb = K / BK;
    for (int kb = 0; kb < nkb; ++kb) {
        const int cur = kb & 1;
        const bool has_next = (kb + 1 < nkb);

        // issue next tile's global loads before compute (latency hidden)
        if (has_next) {
            ld_a<TA>(A, lda, m0, (kb + 1) * BK, tid, ar);
            ld_b<TB>(Bm, ldb, n0, (kb + 1) * BK, tid, br);
        }

        // ---- compute: 8 k-steps of 4, 8 WMMAs each ----
        const float* __restrict__ Ac = Al[cur];
        const float* __restrict__ Bc = Bl[cur];
#pragma unroll
        for (int kk = 0; kk < BK; kk += 4) {
            const int kidx = kk + half * 2;
            v2f af[4], bf[2];
#pragma unroll
            for (int mi = 0; mi < 4; ++mi)
                af[mi] = *(const v2f*)&Ac[(wm + mi * 16 + l15) * AS + kidx];
#pragma unroll
            for (int ni = 0; ni < 2; ++ni)
                bf[ni] = *(const v2f*)&Bc[(wn + ni * 16 + l15) * BS + kidx];
#pragma unroll
            for (int mi = 0; mi < 4; ++mi)
#pragma unroll
                for (int ni = 0; ni < 2; ++ni)
                    acc[mi][ni] = __builtin_amdgcn_wmma_f32_16x16x4_f32(
                        false, af[mi], false, bf[ni],
                        (short)0, acc[mi][ni], false, false);
        }

        // drain prefetched registers into the alternate buffer
        if (has_next) {
            st_a<TA>(Al[cur ^ 1], tid, ar);
            st_b<TB>(Bl[cur ^ 1], tid, br);
        }
        __syncthreads();
    }

    // ---- epilogue + store (C layout: VGPR r, lanes0-15 -> M=r, lanes16-31 -> M=r+8)
#pragma unroll
    for (int mi = 0; mi < 4; ++mi) {
#pragma unroll
        for (int ni = 0; ni < 2; ++ni) {
            const int col = n0 + wn + ni * 16 + l15;
#pragma unroll
            for (int r = 0; r < 8; ++r) {
                const int row = m0 + wm + mi * 16 + half * 8 + r;
                float v = acc[mi][ni][r];
                if (EPI == EPI_SCALE) {
                    v *= STEP;
                } else if (EPI == EPI_S) {
                    v = ((row == col) ? 1.0f : 0.0f) - STEP * v;
                } else if (EPI == EPI_SOFT) {
                    v = soft_thresh(v + extra[(size_t)row * ldc + col]);
                }
                C[(size_t)row * ldc + col] = v;
            }
        }
    }
}

// z1 = soft(WTx): first layer has Z=0 so the GEMM collapses to elementwise.
__global__ void first_layer_kernel(const float* __restrict__ wtx,
                                   float* __restrict__ z, int n) {
    int i = blockIdx.x * blockDim.x + threadIdx.x;
    if (i < n) z[i] = soft_thresh(wtx[i]);
}

extern "C" void kernel_launch(void* const* d_in, const int* in_sizes, int n_in,
                              void* d_out, int out_size, void* d_ws, size_t ws_size,
                              hipStream_t stream) {
    (void)in_sizes; (void)n_in; (void)out_size; (void)ws_size;

    const float* x = (const float*)d_in[0];   // [B, M, 1] == [B, M]
    const float* W = (const float*)d_in[1];   // [M, P]

    float* xhat = (float*)d_out;                       // [B, M]
    float* zA   = (float*)d_out + (size_t)Bdim * Mdim; // zT slot of d_out, [B, P]

    float* S   = (float*)d_ws;                         // [P, P]
    float* WTx = S + (size_t)Pdim * Pdim;              // [B, P]
    float* zB  = WTx + (size_t)Bdim * Pdim;            // [B, P]

    const dim3 blk(256);

    // 1) S = I - 0.1 * W^T W    (rows/cols = P, K = M, A = W accessed transposed)
    gemm_wmma<1, 0, EPI_S><<<dim3(Pdim / BN, Pdim / BM), blk, 0, stream>>>(
        W, Pdim, W, Pdim, S, Pdim, nullptr, Mdim);

    // 2) WTx = 0.1 * X W        (rows = B, cols = P, K = M)
    gemm_wmma<0, 0, EPI_SCALE><<<dim3(Pdim / BN, Bdim / BM), blk, 0, stream>>>(
        x, Mdim, W, Pdim, WTx, Pdim, nullptr, Mdim);

    // 3) z1 = soft(WTx)  -> zB   (layer 1 with Z=0)
    {
        const int n = Bdim * Pdim;
        first_layer_kernel<<<(n + 255) / 256, 256, 0, stream>>>(WTx, zB, n);
    }

    // 4) layers 2..30: Z <- soft(Z S + WTx); parity puts z30 into zA (= d_out zT)
    for (int t = 2; t <= NUM_LAYERS; ++t) {
        const float* zin  = (t % 2 == 0) ? zB : zA;
        float*       zout = (t % 2 == 0) ? zA : zB;
        gemm_wmma<0, 0, EPI_SOFT><<<dim3(Pdim / BN, Bdim / BM), blk, 0, stream>>>(
            zin, Pdim, S, Pdim, zout, Pdim, WTx, Pdim);
    }

    // 5) xhat = Z W^T           (rows = B, cols = M, K = P, B = W accessed transposed)
    gemm_wmma<0, 1, EPI_NONE><<<dim3(Mdim / BN, Bdim / BM), blk, 0, stream>>>(
        zA, Pdim, W, Pdim, xhat, Mdim, nullptr, Pdim);
}